// TransE_8564164788313
// MI455X (gfx1250) — compile-verified
//
#include <hip/hip_runtime.h>
#include <hip/hip_bf16.h>
#include <stdint.h>

typedef __attribute__((ext_vector_type(2))) float v2f;
typedef __attribute__((ext_vector_type(8))) float v8f;

#define DIMD 128
#define L1EPS 1e-12f

// Fast reciprocal: v_rcp_f32 (~1 ulp) + one Newton-Raphson step (~0.5 ulp).
// Avoids the 12-op IEEE div expansion in the WMMA epilogue.
__device__ __forceinline__ float fast_rcp(float x) {
    float r = __builtin_amdgcn_rcpf(x);
    return fmaf(r, fmaf(-x, r, 1.0f), r);
}

// ---------------------------------------------------------------------------
// Kernel 1: rnorm[n] = 1 / max(sum_d |z[n,d]|, eps)
// One wave handles a 16-row tile of z. A = |z| tile (16x4 f32), B = ones
// (4x16), accumulated over K=128 with V_WMMA_F32_16X16X4_F32. Every output
// column of the 16x16 accumulator equals the row norms, so lanes 0 and 16
// read 8 norms each straight out of the accumulator VGPRs (C/D layout:
// VGPR v -> M=v (lanes 0-15), M=v+8 (lanes 16-31)).
// A layout (32-bit 16x4): VGPR0 = K=0 (lanes 0-15) / K=2 (lanes 16-31),
// VGPR1 = K=1 / K=3  => lane (half,lr) supplies z[row][k0+2*half .. +1].
// ---------------------------------------------------------------------------
__global__ __launch_bounds__(256) void k_norms_wmma(const float* __restrict__ z,
                                                    float* __restrict__ rnorm,
                                                    int N, int ntiles) {
    const int lane = threadIdx.x & 31;
    const int wave = blockIdx.x * (blockDim.x >> 5) + (threadIdx.x >> 5);
    if (wave >= ntiles) return;                 // uniform per wave: EXEC stays all-1s

    const int base = wave << 4;
    const int half = lane >> 4;
    const int lr   = lane & 15;
    int row = base + lr;
    if (row >= N) row = N - 1;                  // clamp loads only; stores guarded below
    const float* p = z + (size_t)row * DIMD + (half << 1);

    v2f b; b[0] = 1.0f; b[1] = 1.0f;            // all-ones B: layout-independent
    v8f acc = {};
    #pragma unroll 4
    for (int k0 = 0; k0 < DIMD; k0 += 4) {
        v2f a;
        a[0] = fabsf(p[k0]);
        a[1] = fabsf(p[k0 + 1]);
        acc = __builtin_amdgcn_wmma_f32_16x16x4_f32(
            /*neg_a=*/false, a, /*neg_b=*/false, b,
            /*c_mod=*/(short)0, acc, /*reuse_a=*/false, /*reuse_b=*/false);
    }

    if (lr == 0) {                              // lanes 0 and 16 hold column N=0
        const int r0 = base + (half << 3);
        #pragma unroll
        for (int v = 0; v < 8; ++v) {
            const int r = r0 + v;
            const float inv = fast_rcp(fmaxf(acc[v], L1EPS));
            if (r < N) rnorm[r] = inv;
        }
    }
}

// ---------------------------------------------------------------------------
// Kernel 2: per-edge score. rel_emb (500*128 f32 = 256000 B) is staged into
// LDS via CDNA5 async copies (ASYNCcnt path), then one wave handles one edge:
// coalesced float4 row reads of head/tail, rel served from LDS, wave32
// shuffle reduction of the L1 distance.
// ---------------------------------------------------------------------------
__global__ __launch_bounds__(1024) void k_score(const float* __restrict__ z,
                                                const int*   __restrict__ eidx,
                                                const int*   __restrict__ etype,
                                                const float* __restrict__ rel,
                                                const float* __restrict__ rnorm,
                                                float* __restrict__ out,
                                                int E, int relFloats) {
    extern __shared__ float lds_rel[];

    // --- stage rel table: global -> LDS, asynchronous (tracked by ASYNCcnt) ---
    {
        // Low 32 bits of a generic LDS pointer are the LDS byte address
        // (ISA 10.2).
        const uint32_t ldsBase = (uint32_t)(uintptr_t)(void*)lds_rel;
        const int nvec = relFloats >> 2;        // float4 chunks
        for (int i = threadIdx.x; i < nvec; i += blockDim.x) {
            const uint32_t ldsb = ldsBase + (uint32_t)i * 16u;
            const uint64_t ga   = (uint64_t)(uintptr_t)(rel + (size_t)i * 4);
            asm volatile("global_load_async_to_lds_b128 %0, %1, off"
                         :: "v"(ldsb), "v"(ga) : "memory");
        }
        asm volatile("s_wait_asynccnt 0" ::: "memory");
    }
    __syncthreads();

    const int lane   = threadIdx.x & 31;
    const int wpb    = blockDim.x >> 5;
    const int gwave  = blockIdx.x * wpb + (threadIdx.x >> 5);
    const int nwaves = gridDim.x * wpb;

    for (int e = gwave; e < E; e += nwaves) {   // uniform trip per wave
        const int eu = __builtin_amdgcn_readfirstlane(e);  // force SGPR -> s_load indices
        const int h = eidx[eu];
        const int t = eidx[E + eu];
        const int r = etype[eu];
        const float sh = rnorm[h];
        const float st = rnorm[t];

        const float4* ph = (const float4*)(z + (size_t)h * DIMD);
        const float4* pt = (const float4*)(z + (size_t)t * DIMD);
        const float4* pr = (const float4*)(lds_rel + (size_t)r * DIMD);

        const float4 a  = ph[lane];             // coalesced 512B row read
        const float4 bb = pt[lane];
        const float4 c  = pr[lane];             // ds_load_b128

        float s;
        s  = fabsf(fmaf(a.x, sh, fmaf(bb.x, -st, c.x)));
        s += fabsf(fmaf(a.y, sh, fmaf(bb.y, -st, c.y)));
        s += fabsf(fmaf(a.z, sh, fmaf(bb.z, -st, c.z)));
        s += fabsf(fmaf(a.w, sh, fmaf(bb.w, -st, c.w)));

        #pragma unroll
        for (int off = 16; off > 0; off >>= 1)
            s += __shfl_xor(s, off, 32);        // wave32 reduction

        if (lane == 0) out[eu] = -s;
    }
}

// ---------------------------------------------------------------------------
extern "C" void kernel_launch(void* const* d_in, const int* in_sizes, int n_in,
                              void* d_out, int out_size, void* d_ws, size_t ws_size,
                              hipStream_t stream) {
    const float* z     = (const float*)d_in[0];   // [N,128] f32
    const int*   eidx  = (const int*)d_in[1];     // [2,E] int32 (per harness int convention)
    const int*   etype = (const int*)d_in[2];     // [E]
    const float* rel   = (const float*)d_in[3];   // [R,128] f32
    float* out   = (float*)d_out;                 // [E]
    float* rnorm = (float*)d_ws;                  // scratch: N*4 bytes (400 KB)

    const int N = in_sizes[0] / DIMD;
    const int E = in_sizes[2];
    const int R = in_sizes[3] / DIMD;

    // Phase 1: reciprocal L1 norms via WMMA (16 rows per wave, 8 waves/block)
    const int ntiles = (N + 15) / 16;
    const int nblk1  = (ntiles + 7) / 8;
    k_norms_wmma<<<nblk1, 256, 0, stream>>>(z, rnorm, N, ntiles);

    // Phase 2: edge scores. 256000 B dynamic LDS (< 320KB/WGP), 1024-thread
    // blocks -> one resident workgroup still yields 8 waves per SIMD32.
    const int relFloats = R * DIMD;
    const size_t ldsBytes = (size_t)relFloats * sizeof(float);
    k_score<<<256, 1024, ldsBytes, stream>>>(z, eidx, etype, rel, rnorm, out,
                                             E, relFloats);
}